// GammaNeuronNet_27195732918355
// MI455X (gfx1250) — compile-verified
//
#include <hip/hip_runtime.h>
#include <math.h>

#define NN 4096
#define IN_LEN 3000
#define BETA 0.125f
#define V_TH -15.0f
#define A_R 1.0f
#define A_D 5.0f

typedef __attribute__((ext_vector_type(2))) float v2f;
typedef __attribute__((ext_vector_type(8))) float v8f;

__device__ __forceinline__ float wave_sum(float v) {
    #pragma unroll
    for (int off = 16; off > 0; off >>= 1)
        v += __shfl_xor(v, off, 32);
    return v;
}

// ---- mean of input_V -------------------------------------------------------
__global__ void k_mean(const float* __restrict__ inV, float* __restrict__ avg) {
    __shared__ float red[8];
    float s = 0.f;
    for (int i = threadIdx.x; i < IN_LEN; i += 256) s += inV[i];
    s = wave_sum(s);
    if ((threadIdx.x & 31) == 0) red[threadIdx.x >> 5] = s;
    __syncthreads();
    if (threadIdx.x == 0) {
        float t = 0.f;
        #pragma unroll
        for (int i = 0; i < 8; ++i) t += red[i];
        *avg = t / (float)IN_LEN;
    }
}

// ---- initial state: V0, s0 = s_inf(V0), sE0; zero accumulators -------------
__global__ void k_init(const float* __restrict__ inV, const float* __restrict__ avg,
                       const float* __restrict__ E_syn,
                       float* __restrict__ V, float* __restrict__ s, float* __restrict__ sE,
                       float* __restrict__ co_syn, float* __restrict__ int_syn,
                       float* __restrict__ int_gap) {
    int i = blockIdx.x * 256 + threadIdx.x;
    if (i >= NN) return;
    float v = (i < IN_LEN) ? inV[i] : *avg;
    float sig = 1.f / (1.f + expf(-(BETA * (v - V_TH))));
    float s0 = (A_R * sig) / (A_R * sig + A_D);
    V[i] = v;
    s[i] = s0;
    sE[i] = s0 * E_syn[i];
    co_syn[i] = 0.f; int_syn[i] = 0.f; int_gap[i] = 0.f;
}

// ---- co_gap = rowsum(G_gap), one wave per row ------------------------------
__global__ void k_rowsum(const float* __restrict__ G, float* __restrict__ out) {
    int row = blockIdx.x * 8 + (threadIdx.x >> 5);
    int lane = threadIdx.x & 31;
    const float* r = G + (size_t)row * NN;
    float s = 0.f;
    for (int c = lane * 4; c < NN; c += 128) {
        float4 v = *(const float4*)(r + c);
        s += (v.x + v.y) + (v.z + v.w);
    }
    s = wave_sum(s);
    if (lane == 0) out[row] = s;
}

// ---- fused matvecs via V_WMMA_F32_16X16X4_F32 ------------------------------
// blocks 0..127:  G_syn,  B col0 = s, col1 = s*E_syn  -> co_syn, int_syn
// blocks 128..255: G_gap, B col0 = V                  -> int_gap
// Each block: 8 waves x 16 rows = 128 rows, one of 4 k-chunks (1024 cols).
__launch_bounds__(256)
__global__ void k_matvec(const float* __restrict__ G_syn, const float* __restrict__ G_gap,
                         const float* __restrict__ sV, const float* __restrict__ sEV,
                         const float* __restrict__ Vv,
                         float* __restrict__ co_syn, float* __restrict__ int_syn,
                         float* __restrict__ int_gap) {
    __shared__ float b0_sm[NN];
    __shared__ float b1_sm[NN];
    const int tid = threadIdx.x;
    const int bid = blockIdx.x;
    const bool isSyn = bid < 128;
    const int b2 = bid & 127;
    const int kchunk = b2 >> 5;    // 0..3
    const int rowWG  = b2 & 31;    // 0..31

    if (isSyn) {
        for (int i = tid; i < NN; i += 256) { b0_sm[i] = sV[i]; b1_sm[i] = sEV[i]; }
    } else {
        for (int i = tid; i < NN; i += 256) { b0_sm[i] = Vv[i]; }
    }
    __syncthreads();

    const int wave  = tid >> 5;
    const int lane  = tid & 31;
    const int n     = lane & 15;   // A row within tile / D column
    const int khalf = lane >> 4;   // 0: k%4 in {0,1}, 1: {2,3}
    const int rblk  = rowWG * 8 + wave;     // 16-row block index (0..255)
    const int row   = rblk * 16 + n;

    const float* G = isSyn ? G_syn : G_gap;
    const float* arow  = G + (size_t)row * NN + 2 * khalf;
    const float* bbase = ((n == 1) ? b1_sm : b0_sm) + 2 * khalf;
    const float  bmask = (n < (isSyn ? 2 : 1)) ? 1.0f : 0.0f;
    const int    k0    = kchunk * 1024;

    v8f acc0 = {}, acc1 = {}, acc2 = {}, acc3 = {};
    for (int k = k0; k < k0 + 1024; k += 16) {
        v2f a0 = *(const v2f*)(arow + k);
        v2f a1 = *(const v2f*)(arow + k + 4);
        v2f a2 = *(const v2f*)(arow + k + 8);
        v2f a3 = *(const v2f*)(arow + k + 12);
        v2f b0 = *(const v2f*)(bbase + k);       b0 *= bmask;
        v2f b1 = *(const v2f*)(bbase + k + 4);   b1 *= bmask;
        v2f b2v = *(const v2f*)(bbase + k + 8);  b2v *= bmask;
        v2f b3 = *(const v2f*)(bbase + k + 12);  b3 *= bmask;
        acc0 = __builtin_amdgcn_wmma_f32_16x16x4_f32(false, a0, false, b0,  (short)0, acc0, false, false);
        acc1 = __builtin_amdgcn_wmma_f32_16x16x4_f32(false, a1, false, b1,  (short)0, acc1, false, false);
        acc2 = __builtin_amdgcn_wmma_f32_16x16x4_f32(false, a2, false, b2v, (short)0, acc2, false, false);
        acc3 = __builtin_amdgcn_wmma_f32_16x16x4_f32(false, a3, false, b3,  (short)0, acc3, false, false);
    }
    v8f acc = (acc0 + acc1) + (acc2 + acc3);

    // D[m][n]: vgpr g holds M = g + 8*khalf, N = n. Columns 0/1 hold results.
    const int mbase = rblk * 16 + khalf * 8;
    if (n == 0) {
        float* out0 = isSyn ? co_syn : int_gap;
        #pragma unroll
        for (int g = 0; g < 8; ++g) atomicAdd(&out0[mbase + g], acc[g]);
    } else if (n == 1 && isSyn) {
        #pragma unroll
        for (int g = 0; g < 8; ++g) atomicAdd(&int_syn[mbase + g], acc[g]);
    }
}

// ---- per-neuron state update; re-zero accumulators for next step -----------
__global__ void k_update(const float* __restrict__ G_leak, const float* __restrict__ E_leak,
                         const float* __restrict__ E_syn, const float* __restrict__ co_gap,
                         float* __restrict__ V, float* __restrict__ s, float* __restrict__ sE,
                         float* __restrict__ co_syn, float* __restrict__ int_syn,
                         float* __restrict__ int_gap, const float* __restrict__ dt_p) {
    int i = blockIdx.x * 256 + threadIdx.x;
    if (i >= NN) return;
    float dt = *dt_p;
    float v = V[i], si = s[i];
    float cs = co_syn[i], isn = int_syn[i], ig = int_gap[i];
    float gl = G_leak[i], el = E_leak[i], cg = co_gap[i];
    float I_leak = gl * (v - el);
    float I_syn  = v * cs - isn;
    float I_gap  = v * cg - ig;
    float dV = -(I_leak + I_syn + I_gap);
    float sig = 1.f / (1.f + expf(-(BETA * (v - V_TH))));
    float ds = A_R * sig * (1.f - si) - A_D * si;
    float V_inf = (gl * el + isn + ig) / (gl + cs + cg);
    float diff = fabsf(V_inf - v);
    float vstep = fminf(fmaxf(dV * dt, -diff), diff);
    float vn = v + vstep;
    float sn = si + ds * dt;
    V[i] = vn;
    s[i] = sn;
    sE[i] = sn * E_syn[i];
    co_syn[i] = 0.f; int_syn[i] = 0.f; int_gap[i] = 0.f;
}

// ---- out = V * mask --------------------------------------------------------
__global__ void k_final(const float* __restrict__ V, float* __restrict__ out) {
    int i = blockIdx.x * 256 + threadIdx.x;
    if (i >= NN) return;
    out[i] = (i < IN_LEN) ? V[i] : 0.0f;
}

extern "C" void kernel_launch(void* const* d_in, const int* in_sizes, int n_in,
                              void* d_out, int out_size, void* d_ws, size_t ws_size,
                              hipStream_t stream) {
    const float* input_V = (const float*)d_in[0];
    const float* G_leak  = (const float*)d_in[1];
    const float* E_leak  = (const float*)d_in[2];
    const float* G_syn   = (const float*)d_in[3];
    const float* E_syn   = (const float*)d_in[4];
    const float* G_gap   = (const float*)d_in[5];
    const float* dt_p    = (const float*)d_in[6];
    // d_in[7] = runtime (static 30.0 per reference)

    float* w       = (float*)d_ws;
    float* avg     = w;            // [1] (+pad to 16)
    float* V       = w + 16;       // [NN]
    float* s       = V + NN;
    float* sE      = s + NN;
    float* co_gap  = sE + NN;
    float* co_syn  = co_gap + NN;
    float* int_syn = co_syn + NN;
    float* int_gap = int_syn + NN;

    // Replicate the reference's float-accumulation step count (static 0.1/30.0).
    int nsteps = 0;
    { double t = 0.0; while (t < 30.0) { t += 0.1; ++nsteps; } }  // -> 300

    k_mean<<<1, 256, 0, stream>>>(input_V, avg);
    k_init<<<NN / 256, 256, 0, stream>>>(input_V, avg, E_syn, V, s, sE,
                                         co_syn, int_syn, int_gap);
    k_rowsum<<<NN / 8, 256, 0, stream>>>(G_gap, co_gap);

    for (int t = 0; t < nsteps; ++t) {
        k_matvec<<<256, 256, 0, stream>>>(G_syn, G_gap, s, sE, V,
                                          co_syn, int_syn, int_gap);
        k_update<<<NN / 256, 256, 0, stream>>>(G_leak, E_leak, E_syn, co_gap,
                                               V, s, sE, co_syn, int_syn, int_gap, dt_p);
    }
    k_final<<<NN / 256, 256, 0, stream>>>(V, (float*)d_out);
}